// SGL_12575664242810
// MI455X (gfx1250) — compile-verified
//
#include <hip/hip_runtime.h>
#include <hip/hip_bf16.h>
#include <stdint.h>

#define N_USER  100000
#define N_ITEM  200000
#define N_NODE  300000
#define LATDIM  64
#define N_LAYER 3
#define CHUNK   256      // edges per staged chunk
#define CPB     4        // chunks per block (software pipeline depth)
#define TPB     256

typedef __attribute__((ext_vector_type(4))) float v4f;

// ---- CDNA5 async global->LDS path (guarded; scalar fallback otherwise) ----
#if defined(__AMDGCN__) && __has_builtin(__builtin_amdgcn_global_load_async_to_lds_b32)
#define HAVE_ASYNC_LDS 1
#else
#define HAVE_ASYNC_LDS 0
#endif

#if HAVE_ASYNC_LDS
#define AS1 __attribute__((address_space(1)))
#define AS3 __attribute__((address_space(3)))
#define CPOL_NT 1   // gfx12+ CPol: TH[2:0], TH_NT=1 -> edge stream won't pollute L2
__device__ __forceinline__ void async_copy_b32(const void* gsrc, void* ldst) {
    __builtin_amdgcn_global_load_async_to_lds_b32(
        (AS1 int*)(gsrc), (AS3 int*)(ldst), 0, CPOL_NT);
}
template <int N> __device__ __forceinline__ void wait_async() {
#if __has_builtin(__builtin_amdgcn_s_wait_asynccnt)
    __builtin_amdgcn_s_wait_asynccnt(N);
#else
    asm volatile("s_wait_asynccnt %0" :: "i"(N) : "memory");
#endif
}
#define WAIT_ASYNC(n) wait_async<n>()
#else
#define WAIT_ASYNC(n) ((void)0)
#endif

struct __align__(8) RC { int r; int c; };

// ---- SpMM: dst[r] += v * src[c]; double-buffered async edge staging ----
// (defined FIRST so the disasm snippet shows this hot kernel)
__global__ __launch_bounds__(TPB) void lgc_spmm(
    const int*   __restrict__ rows,
    const int*   __restrict__ cols,
    const float* __restrict__ vals,
    const float* __restrict__ src,
    float*       __restrict__ dst,
    int nEdges) {
    __shared__ RC    s_rc[2][CHUNK];
    __shared__ float s_v[2][CHUNK];

    const int tid       = threadIdx.x;
    const int blockBase = blockIdx.x * (CHUNK * CPB);
    if (blockBase >= nEdges) return;
    const int rem     = nEdges - blockBase;
    const int nChunks = (rem < CHUNK * CPB) ? (rem + CHUNK - 1) / CHUNK : CPB;

    auto stage = [&](int buf, int sbase) {
        const int e = sbase + tid;
        if (sbase + CHUNK <= nEdges) {
#if HAVE_ASYNC_LDS
            // per-lane async DMA straight into LDS (ASYNCcnt-tracked, NT hint)
            async_copy_b32(rows + e, &s_rc[buf][tid].r);
            async_copy_b32(cols + e, &s_rc[buf][tid].c);
            async_copy_b32(vals + e, &s_v[buf][tid]);
#else
            s_rc[buf][tid].r = __builtin_nontemporal_load(rows + e);
            s_rc[buf][tid].c = __builtin_nontemporal_load(cols + e);
            s_v[buf][tid]    = __builtin_nontemporal_load(vals + e);
#endif
        } else {
            const bool inb = (e < nEdges);
            s_rc[buf][tid].r = inb ? rows[e] : 0;
            s_rc[buf][tid].c = inb ? cols[e] : 0;
            s_v[buf][tid]    = inb ? vals[e] : 0.f;
        }
    };

    stage(0, blockBase);                       // prologue: chunk 0

    const int lane = tid & 15;                 // which float4 of the 64-float row
    const int sub  = tid >> 4;                 // which edge within a group of 16

    for (int k = 0; k < nChunks; ++k) {
        const int nextBase = blockBase + (k + 1) * CHUNK;
        const bool haveNext = (k + 1 < nChunks);
        // nextAsync: next stage will put exactly 3 ops on ASYNCcnt
        const bool nextAsync = haveNext && (nextBase + CHUNK <= nEdges);
        if (haveNext) stage((k + 1) & 1, nextBase);
        if (nextAsync) { WAIT_ASYNC(3); } else { WAIT_ASYNC(0); }
        __syncthreads();                       // chunk k fully staged, all waves

        const int b    = k & 1;
        const int base = blockBase + k * CHUNK;
#pragma unroll 4
        for (int i = 0; i < CHUNK / 16; ++i) {
            const int le = sub + i * 16;       // local edge id, monotone in i
            if (base + le >= nEdges) break;
            const RC    rc = s_rc[b][le];      // one ds_load_b64 (broadcast)
            const float v  = s_v[b][le];       // one ds_load_b32 (broadcast)
            // 32-bit offsets: col*64 <= 19.2M, row*64+63 <= 19.2M -> fits easily
            const v4f m = ((const v4f*)(src + rc.c * LATDIM))[lane]; // 256B/row
            float* dp = dst + rc.r * LATDIM + lane * 4;
            unsafeAtomicAdd(dp + 0, v * m.x);  // native global_atomic_add_f32
            unsafeAtomicAdd(dp + 1, v * m.y);
            unsafeAtomicAdd(dp + 2, v * m.z);
            unsafeAtomicAdd(dp + 3, v * m.w);
        }
        __syncthreads();                       // buf b free before iter k+1 restages it
    }
}

// ---- init: A = concat(u,i) (RT: keep in L2 for gather); acc = same (NT stream) ----
__global__ void lgc_init(const v4f* __restrict__ u, const v4f* __restrict__ it,
                         v4f* __restrict__ a, v4f* __restrict__ acc) {
    const int i = blockIdx.x * TPB + threadIdx.x;   // < 4.8M, fits in int
    const int nu = N_USER * (LATDIM / 4);
    const int nt = N_NODE * (LATDIM / 4);
    if (i >= nt) return;
    v4f v;
    if (i < nu) v = __builtin_nontemporal_load(u + i);
    else        v = __builtin_nontemporal_load(it + (i - nu));
    a[i] = v;                              // regular store: resident for gathers
    __builtin_nontemporal_store(v, acc + i);
}

// zero with regular stores: these lines are about to take 640M atomics in L2
__global__ void lgc_zero(v4f* __restrict__ dst) {
    const int i = blockIdx.x * TPB + threadIdx.x;
    const int nt = N_NODE * (LATDIM / 4);
    if (i >= nt) return;
    dst[i] = (v4f)(0.f);
}

// acc (d_out) is a pure stream: NT both ways; x is next layer's gather source: RT read
__global__ void lgc_acc(v4f* __restrict__ acc, const v4f* __restrict__ x) {
    const int i = blockIdx.x * TPB + threadIdx.x;
    const int nt = N_NODE * (LATDIM / 4);
    if (i >= nt) return;
    v4f a = __builtin_nontemporal_load(acc + i);
    v4f b = x[i];
    __builtin_nontemporal_store(a + b, acc + i);
}

extern "C" void kernel_launch(void* const* d_in, const int* in_sizes, int n_in,
                              void* d_out, int out_size, void* d_ws, size_t ws_size,
                              hipStream_t stream) {
    const int*   rows = (const int*)d_in[0];
    const int*   cols = (const int*)d_in[1];
    const float* vals = (const float*)d_in[2];
    const v4f*   u    = (const v4f*)d_in[3];
    const v4f*   it   = (const v4f*)d_in[4];
    float* acc = (float*)d_out;
    const int nEdges = in_sizes[0];

    // ws: two ping-pong node buffers, 300000*64 floats each (76.8 MB)
    float* A = (float*)d_ws;
    float* B = A + (size_t)N_NODE * LATDIM;

    const int nt4   = N_NODE * (LATDIM / 4);                 // 4.8M float4s
    const int gElem = (nt4 + TPB - 1) / TPB;
    const int gEdge = (nEdges + CHUNK * CPB - 1) / (CHUNK * CPB);

    lgc_init<<<gElem, TPB, 0, stream>>>(u, it, (v4f*)A, (v4f*)acc);

    float* in_l  = A;
    float* out_l = B;
    for (int l = 0; l < N_LAYER; ++l) {
        lgc_zero<<<gElem, TPB, 0, stream>>>((v4f*)out_l);
        lgc_spmm<<<gEdge, TPB, 0, stream>>>(rows, cols, vals, in_l, out_l, nEdges);
        lgc_acc<<<gElem, TPB, 0, stream>>>((v4f*)acc, (const v4f*)out_l);
        float* t = in_l; in_l = out_l; out_l = t;
    }
}